// GCNRegressor_78443282694679
// MI455X (gfx1250) — compile-verified
//
#include <hip/hip_runtime.h>
#include <hip/hip_bf16.h>

#define DH 128

typedef __attribute__((ext_vector_type(2))) float v2f;
typedef __attribute__((ext_vector_type(8))) float v8f;

static __device__ __forceinline__ float relu_f(float v) { return v > 0.0f ? v : 0.0f; }

// ---------------- normalization ----------------

__global__ __launch_bounds__(256) void k_init_deg(float* deg, int n) {
    int i = blockIdx.x * 256 + threadIdx.x;
    if (i < n) deg[i] = 1.0f;  // self-loop weight
}

__global__ __launch_bounds__(256) void k_deg_scatter(const int* __restrict__ dst,
                                                     const float* __restrict__ w,
                                                     float* deg, int e) {
    int i = blockIdx.x * 256 + threadIdx.x;
    if (i < e) unsafeAtomicAdd(&deg[dst[i]], w[i]);
}

__global__ __launch_bounds__(256) void k_dis(const float* __restrict__ deg,
                                             float* __restrict__ dis,
                                             float* __restrict__ selfnorm, int n) {
    int i = blockIdx.x * 256 + threadIdx.x;
    if (i < n) {
        float d = deg[i];
        float r = d > 0.0f ? rsqrtf(d) : 0.0f;
        dis[i] = r;
        selfnorm[i] = r * r;  // dis[i] * 1.0 * dis[i] (self-loop norm)
    }
}

__global__ __launch_bounds__(256) void k_norm(const int* __restrict__ src,
                                              const int* __restrict__ dst,
                                              const float* __restrict__ w,
                                              const float* __restrict__ dis,
                                              float* __restrict__ norm, int e) {
    int i = blockIdx.x * 256 + threadIdx.x;
    if (i < e) norm[i] = dis[src[i]] * w[i] * dis[dst[i]];
}

// ---------------- dense GEMM: H[n,128] = X[n,128] @ W[128,128] ----------------
// Block = 256 threads = 8 waves. Each block computes a 16-row strip; wave w
// owns the 16x16 tile at columns [16w, 16w+16). K loop: 32 x V_WMMA_F32_16X16X4_F32.
// W is staged into LDS once per block (64 KB of the 320 KB WGP LDS).

__global__ __launch_bounds__(256) void k_gemm128(const float* __restrict__ X,
                                                 const float* __restrict__ W,
                                                 float* __restrict__ H, int n) {
    __shared__ float sW[DH * DH];  // 64 KB
    for (int i = threadIdx.x; i < DH * DH; i += 256) sW[i] = W[i];
    __syncthreads();

    const int wave = threadIdx.x >> 5;
    const int lane = threadIdx.x & 31;
    const int row0 = blockIdx.x * 16;
    const int col0 = wave * 16;
    const int m = lane & 15;        // A: row within tile / B,C,D: column within tile
    const int half = lane >> 4;     // upper half-wave handles K+2 (A/B) or M+8 (C/D)

    int arow = row0 + m;
    if (arow >= n) arow = n - 1;  // clamp (n==100000 is divisible by 16; safety only)
    const float* xrow = X + (size_t)arow * DH;

    v8f c = {};
#pragma unroll
    for (int k0 = 0; k0 < DH; k0 += 4) {
        const int ka = k0 + 2 * half;
        v2f a;
        a.x = xrow[ka];
        a.y = xrow[ka + 1];
        v2f b;
        b.x = sW[ka * DH + col0 + m];
        b.y = sW[(ka + 1) * DH + col0 + m];
        // D = A(16x4) * B(4x16) + C  -> v_wmma_f32_16x16x4_f32
        c = __builtin_amdgcn_wmma_f32_16x16x4_f32(false, a, false, b, (short)0, c,
                                                  false, false);
    }

    const int moff = half * 8;
#pragma unroll
    for (int r = 0; r < 8; ++r) {
        int orow = row0 + moff + r;
        if (orow < n) H[(size_t)orow * DH + col0 + m] = c[r];
    }
}

// ---------------- propagation ----------------

// agg[i,:] = selfnorm[i] * h[i,:]   (self-loop contribution; also zero-inits agg)
__global__ __launch_bounds__(256) void k_init_agg(const float* __restrict__ h,
                                                  const float* __restrict__ selfnorm,
                                                  float* __restrict__ agg, int n) {
    int idx = blockIdx.x * 256 + threadIdx.x;  // one float4 per thread, 32 per row
    int total = n * (DH / 4);
    if (idx < total) {
        int row = idx >> 5;
        float s = selfnorm[row];
        float4 v = ((const float4*)h)[idx];
        float4 o;
        o.x = s * v.x; o.y = s * v.y; o.z = s * v.z; o.w = s * v.w;
        ((float4*)agg)[idx] = o;
    }
}

// One wave per edge: coalesced float4 gather of the 128-wide row, scaled
// scatter-add via hardware global_atomic_add_f32.
__global__ __launch_bounds__(256) void k_scatter(const float* __restrict__ h,
                                                 const int* __restrict__ src,
                                                 const int* __restrict__ dst,
                                                 const float* __restrict__ norm,
                                                 float* __restrict__ agg, int e) {
    int gw = (blockIdx.x * 256 + threadIdx.x) >> 5;  // edge id
    int lane = threadIdx.x & 31;
    if (gw >= e) return;
    int s = src[gw];
    int d = dst[gw];
    float w = norm[gw];
    float4 v = ((const float4*)(h + (size_t)s * DH))[lane];
    float* a = agg + (size_t)d * DH + lane * 4;
    unsafeAtomicAdd(a + 0, w * v.x);
    unsafeAtomicAdd(a + 1, w * v.y);
    unsafeAtomicAdd(a + 2, w * v.z);
    unsafeAtomicAdd(a + 3, w * v.w);
}

// in-place: buf[i,j] = relu(buf[i,j] + b[j])
__global__ __launch_bounds__(256) void k_bias_relu(float* __restrict__ buf,
                                                   const float* __restrict__ b, int n) {
    int idx = blockIdx.x * 256 + threadIdx.x;
    if (idx < n * DH) buf[idx] = relu_f(buf[idx] + b[idx & (DH - 1)]);
}

// ---------------- layer 3 (D_out = 1) ----------------

// h3[i] = dot(h[i,:], W3[:,0])   (one wave per node, shuffle reduction)
__global__ __launch_bounds__(256) void k_dot_w3(const float* __restrict__ h,
                                                const float* __restrict__ W3,
                                                float* __restrict__ h3, int n) {
    int node = (blockIdx.x * 256 + threadIdx.x) >> 5;
    int lane = threadIdx.x & 31;
    if (node >= n) return;
    float4 a = ((const float4*)(h + (size_t)node * DH))[lane];
    float4 b = ((const float4*)W3)[lane];
    float sum = a.x * b.x + a.y * b.y + a.z * b.z + a.w * b.w;
#pragma unroll
    for (int off = 16; off > 0; off >>= 1) sum += __shfl_down(sum, off, 32);
    if (lane == 0) h3[node] = sum;
}

__global__ __launch_bounds__(256) void k_init_agg3(const float* __restrict__ h3,
                                                   const float* __restrict__ selfnorm,
                                                   float* __restrict__ agg3, int n) {
    int i = blockIdx.x * 256 + threadIdx.x;
    if (i < n) agg3[i] = selfnorm[i] * h3[i];
}

__global__ __launch_bounds__(256) void k_scatter3(const float* __restrict__ h3,
                                                  const int* __restrict__ src,
                                                  const int* __restrict__ dst,
                                                  const float* __restrict__ norm,
                                                  float* __restrict__ agg3, int e) {
    int i = blockIdx.x * 256 + threadIdx.x;
    if (i < e) unsafeAtomicAdd(&agg3[dst[i]], norm[i] * h3[src[i]]);
}

__global__ __launch_bounds__(256) void k_final(const float* __restrict__ agg3,
                                               const float* __restrict__ b3,
                                               float* __restrict__ out, int n) {
    int i = blockIdx.x * 256 + threadIdx.x;
    if (i < n) out[i] = relu_f(agg3[i] + b3[0]);
}

// ---------------- host ----------------

static inline int cdiv(long a, long b) { return (int)((a + b - 1) / b); }

extern "C" void kernel_launch(void* const* d_in, const int* in_sizes, int n_in,
                              void* d_out, int out_size, void* d_ws, size_t ws_size,
                              hipStream_t stream) {
    const float* x  = (const float*)d_in[0];
    const int*   ei = (const int*)d_in[1];
    const float* ew = (const float*)d_in[2];
    const float* W1 = (const float*)d_in[3];
    const float* b1 = (const float*)d_in[4];
    const float* W2 = (const float*)d_in[5];
    const float* b2 = (const float*)d_in[6];
    const float* W3 = (const float*)d_in[7];
    const float* b3 = (const float*)d_in[8];
    float* out = (float*)d_out;

    const int n = in_sizes[0] / DH;
    const int e = in_sizes[2];
    const int* src = ei;       // edge_index[0, :]
    const int* dst = ei + e;   // edge_index[1, :]

    // workspace layout (floats)
    float* ws = (float*)d_ws;
    float* deg      = ws;                     // n
    float* dis      = deg + n;                // n
    float* selfnorm = dis + n;                // n
    float* normv    = selfnorm + n;           // e
    float* bufA     = normv + e;              // n*DH
    float* bufB     = bufA + (size_t)n * DH;  // n*DH
    float* h3       = bufB + (size_t)n * DH;  // n
    float* agg3     = h3 + n;                 // n

    // ---- normalization ----
    k_init_deg<<<cdiv(n, 256), 256, 0, stream>>>(deg, n);
    k_deg_scatter<<<cdiv(e, 256), 256, 0, stream>>>(dst, ew, deg, e);
    k_dis<<<cdiv(n, 256), 256, 0, stream>>>(deg, dis, selfnorm, n);
    k_norm<<<cdiv(e, 256), 256, 0, stream>>>(src, dst, ew, dis, normv, e);

    const int gemm_blocks = cdiv(n, 16);
    const int row4_blocks = cdiv((long)n * (DH / 4), 256);
    const int elem_blocks = cdiv((long)n * DH, 256);
    const int edge_wave_blocks = cdiv((long)e * 32, 256);
    const int node_wave_blocks = cdiv((long)n * 32, 256);

    // ---- layer 1: h1 = relu(prop(x @ W1) + b1) ----
    k_gemm128<<<gemm_blocks, 256, 0, stream>>>(x, W1, bufA, n);
    k_init_agg<<<row4_blocks, 256, 0, stream>>>(bufA, selfnorm, bufB, n);
    k_scatter<<<edge_wave_blocks, 256, 0, stream>>>(bufA, src, dst, normv, bufB, e);
    k_bias_relu<<<elem_blocks, 256, 0, stream>>>(bufB, b1, n);  // h1 = bufB

    // ---- layer 2: h2 = relu(prop(h1 @ W2) + b2) ----
    k_gemm128<<<gemm_blocks, 256, 0, stream>>>(bufB, W2, bufA, n);
    k_init_agg<<<row4_blocks, 256, 0, stream>>>(bufA, selfnorm, bufB, n);
    k_scatter<<<edge_wave_blocks, 256, 0, stream>>>(bufA, src, dst, normv, bufB, e);
    k_bias_relu<<<elem_blocks, 256, 0, stream>>>(bufB, b2, n);  // h2 = bufB

    // ---- layer 3: out = relu(prop(h2 @ W3) + b3) ----
    k_dot_w3<<<node_wave_blocks, 256, 0, stream>>>(bufB, W3, h3, n);
    k_init_agg3<<<cdiv(n, 256), 256, 0, stream>>>(h3, selfnorm, agg3, n);
    k_scatter3<<<cdiv(e, 256), 256, 0, stream>>>(h3, src, dst, normv, agg3, e);
    k_final<<<cdiv(n, 256), 256, 0, stream>>>(agg3, b3, out, n);
}